// FMLayer_45595372814827
// MI455X (gfx1250) — compile-verified
//
#include <hip/hip_runtime.h>
#include <hip/hip_bf16.h>

typedef __attribute__((ext_vector_type(2))) float v2f;
typedef __attribute__((ext_vector_type(8))) float v8f;

#define FM_B      8192
#define FEAT      200
#define NUMS      100
#define INCOLS    300
#define INSTRIDE  305   // odd stride -> conflict-free LDS column reads
#define EMB       64
#define TILE_ROWS 16
#define BLOCK     256

__global__ __launch_bounds__(BLOCK) void fm_layer_kernel(
    const float* __restrict__ inputs,     // (B, 300)
    const float* __restrict__ w_one_hot,  // (1e6, 1)
    const float* __restrict__ w_numeric,  // (100, 1)
    const float* __restrict__ v_one_hot,  // (1e6, 64)
    const float* __restrict__ v_numeric,  // (100, 64)
    const float* __restrict__ bias,       // (1,)
    float* __restrict__ out)              // (B, 1)
{
    __shared__ float s_in[TILE_ROWS][INSTRIDE];   // input tile (idx + numeric)
    __shared__ float s_fs[TILE_ROWS][EMB];        // field sums
    __shared__ float s_fss[TILE_ROWS][EMB];       // field square sums
    __shared__ float s_first[TILE_ROWS];          // first-order terms

    const int tid  = threadIdx.x;
    const int lane = tid & 31;
    const int wave = tid >> 5;
    const int row0 = blockIdx.x * TILE_ROWS;

    // ---- Phase 0: stage the 16x300 input tile into LDS (coalesced) ----
    for (int i = tid; i < TILE_ROWS * INCOLS; i += BLOCK) {
        int r = i / INCOLS, c = i % INCOLS;
        s_in[r][c] = inputs[(row0 + r) * INCOLS + c];
    }
    __syncthreads();

    // ---- Phase 1: sparse gathers. wave w handles rows 2w, 2w+1. ----
    // Each lane owns embed elements {2*lane, 2*lane+1} -> one float2 (b64) load
    // per feature, so a wave pulls the full 256B embedding row coalesced.
    const float2* vtab = reinterpret_cast<const float2*>(v_one_hot);
    #pragma unroll
    for (int rr = 0; rr < 2; ++rr) {
        const int r = wave * 2 + rr;
        float fs0 = 0.f, fs1 = 0.f, fss0 = 0.f, fss1 = 0.f;
        float first = 0.f;

        // first-order: sparse weights (lane-parallel over features)
        for (int j = lane; j < FEAT; j += 32) {
            int idx = (int)s_in[r][j];
            first += w_one_hot[idx];
        }
        // first-order: numeric dot product
        for (int k = lane; k < NUMS; k += 32) {
            first += s_in[r][FEAT + k] * w_numeric[k];
        }
        // second-order embedding gather (idx broadcast from LDS)
        #pragma unroll 4
        for (int j = 0; j < FEAT; ++j) {
            int idx = (int)s_in[r][j];
            float2 v = vtab[(size_t)idx * (EMB / 2) + lane];
            fs0 += v.x;  fss0 += v.x * v.x;
            fs1 += v.y;  fss1 += v.y * v.y;
        }
        // wave-reduce first-order term
        for (int off = 16; off > 0; off >>= 1) first += __shfl_xor(first, off, 32);
        if (lane == 0) s_first[r] = first;

        s_fs[r][2 * lane]      = fs0;
        s_fs[r][2 * lane + 1]  = fs1;
        s_fss[r][2 * lane]     = fss0;
        s_fss[r][2 * lane + 1] = fss1;
    }
    __syncthreads();

    // ---- Phase 2: dense numeric part via V_WMMA_F32_16X16X4_F32. ----
    // Waves 0..3 each own a 16-wide N tile. K = 100 -> 25 wmma steps, done
    // twice (plain operands -> field_sum, squared operands -> field_sq_sum).
    if (wave < 4) {
        const int n0    = wave * 16;
        const int m     = lane & 15;            // A: M index / B,C,D: N index
        const int khalf = (lane < 16) ? 0 : 2;  // lanes 16-31 hold K+2, K+3
        v8f acc_fs  = {};
        v8f acc_fss = {};
        for (int kk = 0; kk < NUMS; kk += 4) {
            v2f a, b;
            a.x = s_in[m][FEAT + kk + khalf];
            a.y = s_in[m][FEAT + kk + khalf + 1];
            b.x = v_numeric[(kk + khalf) * EMB + n0 + m];
            b.y = v_numeric[(kk + khalf + 1) * EMB + n0 + m];
            v2f a2 = a * a;
            v2f b2 = b * b;
            acc_fs = __builtin_amdgcn_wmma_f32_16x16x4_f32(
                false, a, false, b, (short)0, acc_fs, false, false);
            acc_fss = __builtin_amdgcn_wmma_f32_16x16x4_f32(
                false, a2, false, b2, (short)0, acc_fss, false, false);
        }
        // C/D layout: VGPR v -> row v (lanes 0-15) or v+8 (lanes 16-31), col = n0+m
        const int rbase = (lane < 16) ? 0 : 8;
        #pragma unroll
        for (int v = 0; v < 8; ++v) {
            s_fs[rbase + v][n0 + m]  += acc_fs[v];
            s_fss[rbase + v][n0 + m] += acc_fss[v];
        }
    }
    __syncthreads();

    // ---- Phase 3: per-row reduction, 16 threads per row ----
    const int g = tid >> 4;   // row within tile
    const int t = tid & 15;
    float p = 0.f;
    #pragma unroll
    for (int e = t; e < EMB; e += 16) {
        float f = s_fs[g][e];
        p += f * f - s_fss[g][e];
    }
    for (int off = 8; off > 0; off >>= 1) p += __shfl_xor(p, off, 16);
    if (t == 0) out[row0 + g] = s_first[g] + 0.5f * p + bias[0];
}

extern "C" void kernel_launch(void* const* d_in, const int* in_sizes, int n_in,
                              void* d_out, int out_size, void* d_ws, size_t ws_size,
                              hipStream_t stream) {
    (void)in_sizes; (void)n_in; (void)d_ws; (void)ws_size; (void)out_size;
    const float* inputs    = (const float*)d_in[0];
    const float* w_one_hot = (const float*)d_in[1];
    const float* w_numeric = (const float*)d_in[2];
    const float* v_one_hot = (const float*)d_in[3];
    const float* v_numeric = (const float*)d_in[4];
    const float* bias      = (const float*)d_in[5];
    float* out             = (float*)d_out;

    dim3 grid(FM_B / TILE_ROWS);   // 512 blocks
    dim3 block(BLOCK);             // 256 threads = 8 waves (wave32)
    fm_layer_kernel<<<grid, block, 0, stream>>>(
        inputs, w_one_hot, w_numeric, v_one_hot, v_numeric, bias, out);
}